// ChamferDistanceMatrixL2_5248450036646
// MI455X (gfx1250) — compile-verified
//
#include <hip/hip_runtime.h>

// Chamfer distance matrix, [B=32, G=64, N=32, C=3] -> out [B, G1, G2] f32.
// Cross products via V_WMMA_F32_16X16X4_F32 (K padded 3 -> 4).
// Reductions via DPP16 row rotations (VALU) instead of ds_bpermute.

typedef float v2f __attribute__((ext_vector_type(2)));
typedef float v8f __attribute__((ext_vector_type(8)));

// DPP16 permute within each 16-lane row; CTRL: row_ror:n = 0x120 + n.
template <int CTRL>
__device__ __forceinline__ float dpp_perm(float x) {
  return __int_as_float(__builtin_amdgcn_update_dpp(
      __float_as_int(x), __float_as_int(x), CTRL, 0xF, 0xF, true));
}

// min across each 16-lane row via rotation butterfly (rot 8,4,2,1) — pure VALU.
__device__ __forceinline__ float row_min16(float v) {
  v = fminf(v, dpp_perm<0x128>(v));  // row_ror:8
  v = fminf(v, dpp_perm<0x124>(v));  // row_ror:4
  v = fminf(v, dpp_perm<0x122>(v));  // row_ror:2
  v = fminf(v, dpp_perm<0x121>(v));  // row_ror:1
  return v;
}

// sum across each 16-lane row via rotation butterfly — pure VALU.
__device__ __forceinline__ float row_sum16(float v) {
  v += dpp_perm<0x128>(v);
  v += dpp_perm<0x124>(v);
  v += dpp_perm<0x122>(v);
  v += dpp_perm<0x121>(v);
  return v;
}

// cross-half exchange (lane <-> lane^16): the one permutation that truly
// needs the lane crossbar on wave32.
__device__ __forceinline__ float xhalf(float v) { return __shfl_xor(v, 16, 32); }

__device__ __forceinline__ float wave_sum32(float v) {
  v = row_sum16(v);
  return v + xhalf(v);
}

// Build the WMMA operand for a 16-point tile. Per the CDNA5 ISA 16x4 f32
// A-layout (and its mirrored 4x16 B-layout): lanes 0-15 hold K=0 (x) in v0 and
// K=1 (y) in v1; lanes 16-31 hold K=2 (z) in v0 and K=3 (pad=0) in v1, with
// the point index = lane%16. The same registers are valid as A of one product
// and B of the transposed product. Also returns the point's squared norm
// (for point tile*16 + lane%16, replicated across both halves).
__device__ __forceinline__ v2f build_tile(const float* __restrict__ grp,
                                          int tile, int l, bool hi, float& sq) {
  const float* p = grp + 3 * (tile * 16 + l);
  float x = p[0], y = p[1], z = p[2];
  sq = fmaf(x, x, fmaf(y, y, z * z));
  v2f op;
  op.x = hi ? z : x;
  op.y = hi ? 0.0f : y;
  return op;
}

// One chamfer direction: c = A x B (32x32 via 4 WMMAs), then
// returns sum over the 32 rows n of min over columns m of (sqB[m] - 2*c[n,m]).
// sqB0/sqB1 are the per-lane squared norms of B's two 16-column tiles
// (column index = lane%16, matching the C/D layout N = lane%16).
__device__ __forceinline__ float chamfer_dir(v2f A0, v2f A1, v2f B0, v2f B1,
                                             float sqB0, float sqB1) {
  v8f zero = {};
  float r = 0.0f;
#pragma unroll
  for (int i = 0; i < 2; ++i) {
    v2f Ai = i ? A1 : A0;
    v8f c0 = __builtin_amdgcn_wmma_f32_16x16x4_f32(
        false, Ai, false, B0, (short)0, zero, false, false);
    v8f c1 = __builtin_amdgcn_wmma_f32_16x16x4_f32(
        false, Ai, false, B1, (short)0, zero, false, false);
    float s = 0.0f;
#pragma unroll
    for (int v = 0; v < 8; ++v) {
      // combine the two column tiles elementwise, then min over the 16-lane row
      float e = fminf(fmaf(-2.0f, c0[v], sqB0), fmaf(-2.0f, c1[v], sqB1));
      e = row_min16(e);             // row-min for row M = v + 8*hi of tile i
      s += e;                       // accumulate 8 row-mins per half
    }
    s += xhalf(s);                  // add the two halves -> 16 rows of tile i
    r += s;
  }
  return r;                         // sum of all 32 row minima
}

__global__ __launch_bounds__(256) void
ChamferDistanceMatrixL2_5248450036646_kernel(const float* __restrict__ xyz1,
                                             const float* __restrict__ xyz2,
                                             float* __restrict__ out) {
  constexpr int G = 64, N = 32, G2CHUNK = 16;

  const int wave = (blockIdx.x * blockDim.x + threadIdx.x) >> 5;
  const int lane = threadIdx.x & 31;
  const int l = lane & 15;
  const bool hi = lane >= 16;

  // wave -> (b, g1, g2 chunk); 8192 waves total
  const int q  = wave & 3;
  const int g1 = (wave >> 2) & 63;
  const int b  = wave >> 8;

  // set-1 group: build once, reuse for all 16 g2 iterations
  const float* grpX = xyz1 + (size_t)((b * G + g1) * N) * 3;
  float sx0, sx1;
  v2f X0 = build_tile(grpX, 0, l, hi, sx0);
  v2f X1 = build_tile(grpX, 1, l, hi, sx1);
  // each x2[n] is counted twice across the wave (both halves) -> /64
  const float meanx2 = wave_sum32(sx0 + sx1) * (1.0f / 64.0f);

  const int g2base = q * G2CHUNK;
  for (int gg = 0; gg < G2CHUNK; ++gg) {
    const int g2 = g2base + gg;
    const float* grpY = xyz2 + (size_t)((b * G + g2) * N) * 3;
    if (gg + 1 < G2CHUNK) {
      __builtin_prefetch(grpY + (size_t)N * 3, 0, 3);  // next g2 group (L2-hot)
    }
    float sy0, sy1;
    v2f Y0 = build_tile(grpY, 0, l, hi, sy0);
    v2f Y1 = build_tile(grpY, 1, l, hi, sy1);
    const float meany2 = wave_sum32(sy0 + sy1) * (1.0f / 64.0f);

    // direction 1: for each x-point, nearest y-point
    const float r1 = chamfer_dir(X0, X1, Y0, Y1, sy0, sy1);
    // direction 2: transposed product, for each y-point, nearest x-point
    const float r2 = chamfer_dir(Y0, Y1, X0, X1, sx0, sx1);

    const float res = meanx2 + meany2 + (r1 + r2) * (1.0f / 32.0f);
    if (lane == 0) out[(size_t)(b * G + g1) * G + g2] = res;
  }
}

extern "C" void kernel_launch(void* const* d_in, const int* in_sizes, int n_in,
                              void* d_out, int out_size, void* d_ws, size_t ws_size,
                              hipStream_t stream) {
  const float* x1 = (const float*)d_in[0];
  const float* x2 = (const float*)d_in[1];
  float* out = (float*)d_out;
  (void)in_sizes; (void)n_in; (void)out_size; (void)d_ws; (void)ws_size;

  // 8192 waves (B*G1*4 g2-chunks), 8 waves per block -> 1024 blocks
  dim3 grid(1024), block(256);
  hipLaunchKernelGGL(ChamferDistanceMatrixL2_5248450036646_kernel,
                     grid, block, 0, stream, x1, x2, out);
}